// FuzzyUserAllocator_2_24472723653392
// MI455X (gfx1250) — compile-verified
//
#include <hip/hip_runtime.h>
#include <stdint.h>

// MI455X / gfx1250, wave32. Sequential greedy allocator: latency-bound scan of
// U=8192 steps; single persistent 256-thread workgroup (8 wave32), all server
// state in registers (4 servers/thread), async LDS double-buffered mask-row
// prefetch (ASYNCcnt path), ballot/popc + wave32 shfl reductions, fast
// rcp/sqrt in the scoring path, readlane broadcasts, and an exact f32 WMMA
// (16x16x4) for the cross-wave sum statistics.

typedef float v2f __attribute__((ext_vector_type(2)));
typedef float v8f __attribute__((ext_vector_type(8)));

#define NTH 256
#define NW  8    // wave32 waves per block
#define SPT 4    // servers per thread (S = 1024)

#if __has_builtin(__builtin_amdgcn_rcpf)
#define FAST_RCP(x) __builtin_amdgcn_rcpf(x)
#else
#define FAST_RCP(x) (1.0f / (x))
#endif
#if __has_builtin(__builtin_amdgcn_sqrtf)
#define FAST_SQRT(x) __builtin_amdgcn_sqrtf(x)
#else
#define FAST_SQRT(x) sqrtf(x)
#endif

__device__ __forceinline__ float readlane_f(float v, int l) {
#if __has_builtin(__builtin_amdgcn_readlane)
    return __int_as_float(__builtin_amdgcn_readlane(__float_as_int(v), l));
#else
    return __shfl(v, l);
#endif
}
__device__ __forceinline__ int readlane_i(int v, int l) {
#if __has_builtin(__builtin_amdgcn_readlane)
    return __builtin_amdgcn_readlane(v, l);
#else
    return __shfl(v, l);
#endif
}

__global__ __launch_bounds__(NTH) void fuzzy_alloc_kernel(
    const float* __restrict__ servers,        // S x 7
    const float* __restrict__ users,          // U x 6
    const unsigned char* __restrict__ masks,  // U x S (bool bytes)
    float* __restrict__ out,                  // [U alloc][S usage][2 props]
    unsigned long long* __restrict__ sortbuf, // ws: U u64 (key<<32 | idx)
    int U, int S)
{
    const int tid  = threadIdx.x;
    const int lane = tid & 31;
    const int wv   = tid >> 5;

    __shared__ float stats[NW][8];
    __shared__ unsigned long long minbuf[NW];
    __shared__ __align__(16) unsigned char maskbuf[2][NTH * SPT];

    // ---------- phase 1: stable ascending argsort of users[:,2] ----------
    // keys in [1,10] -> positive floats, bit pattern is order-preserving;
    // index in the low bits reproduces stable argsort tie-breaking.
    for (int i = tid; i < U; i += NTH) {
        unsigned kb = __float_as_uint(users[i * 6 + 2]);
        sortbuf[i] = ((unsigned long long)kb << 32) | (unsigned)i;
    }
    __syncthreads();
    for (int k = 2; k <= U; k <<= 1) {
        for (int j = k >> 1; j > 0; j >>= 1) {
            for (int i = tid; i < U; i += NTH) {
                int ixj = i ^ j;
                if (ixj > i) {
                    unsigned long long a = sortbuf[i], b = sortbuf[ixj];
                    bool up = ((i & k) == 0);
                    if ((a > b) == up) { sortbuf[i] = b; sortbuf[ixj] = a; }
                }
            }
            __syncthreads();
        }
    }

    // ---------- phase 2: per-server register state ----------
    // tri() divisors replaced by exact-constant reciprocals (1-ulp class).
    const float rmid[4]  = {1.0f / 0.4f, 1.0f / 0.5f, 1.0f / 0.3f, 1.0f / 0.5f};
    const float rspan[4] = {1.0f / 0.4f, 1.0f / 0.3f, 1.0f / 0.4f, 1.0f / 0.3f};
    const float mids[4]  = {0.4f, 0.5f, 0.3f, 0.5f};
    const float highs[4] = {0.8f, 0.8f, 0.7f, 0.8f};
    float cap[SPT][4], rcap[SPT][4], tmp[SPT][4];
    float usage[SPT], fsum[SPT], fsq[SPT], Bv[SPT], Cv[SPT];
#pragma unroll
    for (int j = 0; j < SPT; ++j) {
        int s = tid * SPT + j;
#pragma unroll
        for (int c = 0; c < 4; ++c) {
            cap[j][c] = servers[s * 7 + 3 + c];
            rcap[j][c] = 1.0f / cap[j][c];   // IEEE, once
            tmp[j][c] = cap[j][c];
        }
        usage[j] = 0.0f; fsum[j] = 0.0f; fsq[j] = 0.0f; Bv[j] = 0.0f; Cv[j] = 10.0f;
    }
    int allocCount = 0;
    const int copiers = S >> 4;  // lanes doing 16B async copies of one mask row

    // preload first sorted user's mask row into buffer 0 (async -> LDS)
    {
        unsigned ouid0 = (unsigned)(sortbuf[0] & 0xffffffffu);
        if (tid < copiers) {
            unsigned lds = (unsigned)(uintptr_t)&maskbuf[0][tid * 16];
            unsigned long long ga =
                (unsigned long long)(uintptr_t)(masks + (size_t)ouid0 * S + tid * 16);
            asm volatile("global_load_async_to_lds_b128 %0, %1, off"
                         :: "v"(lds), "v"(ga) : "memory");
        }
        asm volatile("s_wait_asynccnt 0" ::: "memory");
        __syncthreads();
    }

    // ---------- phase 3: sequential greedy scan ----------
    for (int t = 0; t < U; ++t) {
        const int p = t & 1;
        unsigned ouid = (unsigned)(sortbuf[t] & 0xffffffffu);
        const float* urow = users + (size_t)ouid * 6;
        float w0 = urow[2], w1 = urow[3], w2 = urow[4], w3 = urow[5];

        // consume current mask row, then immediately kick the async prefetch
        // of the NEXT sorted user's row into the other buffer (max overlap).
        unsigned mword = *(const unsigned*)&maskbuf[p][tid * 4];
        if (t + 1 < U) {
            unsigned nuid = (unsigned)(sortbuf[t + 1] & 0xffffffffu);
            __builtin_prefetch(users + (size_t)nuid * 6, 0, 0);
            if (tid < copiers) {
                unsigned lds = (unsigned)(uintptr_t)&maskbuf[p ^ 1][tid * 16];
                unsigned long long ga =
                    (unsigned long long)(uintptr_t)(masks + (size_t)nuid * S + tid * 16);
                asm volatile("global_load_async_to_lds_b128 %0, %1, off"
                             :: "v"(lds), "v"(ga) : "memory");
            }
        }

        // local stats over this thread's 4 servers
        float s1 = 0.f, s2 = 0.f;
        float bmx = -INFINITY, bmn = INFINITY;
        bool vld[SPT];
#pragma unroll
        for (int j = 0; j < SPT; ++j) {
            bool v = ((mword >> (8 * j)) & 0xffu) != 0u;
            v = v && (tmp[j][0] >= w0) && (tmp[j][1] >= w1) &&
                     (tmp[j][2] >= w2) && (tmp[j][3] >= w3);
            vld[j] = v;
            if (v) {
                s1 += fsum[j]; s2 += fsq[j];
                bmx = fmaxf(bmx, Bv[j]); bmn = fminf(bmn, Bv[j]);
            }
        }
        // count via ballot/popc (pure scalar, no LDS tree)
        int cnti = __popc((unsigned)__ballot(vld[0])) + __popc((unsigned)__ballot(vld[1])) +
                   __popc((unsigned)__ballot(vld[2])) + __popc((unsigned)__ballot(vld[3]));
        // C in {0,10}: min/max reduce to two "any" ballots
        bool pUnused = (vld[0] && Cv[0] > 0.f) || (vld[1] && Cv[1] > 0.f) ||
                       (vld[2] && Cv[2] > 0.f) || (vld[3] && Cv[3] > 0.f);
        bool pUsed   = (vld[0] && Cv[0] == 0.f) || (vld[1] && Cv[1] == 0.f) ||
                       (vld[2] && Cv[2] == 0.f) || (vld[3] && Cv[3] == 0.f);
        bool anyUnused = __any(pUnused) != 0;
        bool anyUsed   = __any(pUsed) != 0;
        // wave32 float reductions: only s1, s2, bmx, bmn need trees now
        for (int m = 16; m > 0; m >>= 1) {
            s1  += __shfl_xor(s1, m);
            s2  += __shfl_xor(s2, m);
            bmx = fmaxf(bmx, __shfl_xor(bmx, m));
            bmn = fminf(bmn, __shfl_xor(bmn, m));
        }
        if (lane == 0) {
            stats[wv][0] = (float)cnti;
            stats[wv][1] = s1;
            stats[wv][2] = s2;
            stats[wv][3] = bmx;
            stats[wv][4] = bmn;
            stats[wv][5] = anyUnused ? 10.f : (cnti ? 0.f : -INFINITY);   // wave Cmax
            stats[wv][6] = anyUsed   ? 0.f  : (cnti ? 10.f : INFINITY);   // wave Cmin
        }
        __syncthreads();  // barrier 1: wave partials visible

        // cross-wave SUMS via exact f32 WMMA (A = ones 16x4, B carries the 8
        // wave partials of {cnt,s1,s2} in columns 0..5). Lanes with cidx>=6
        // load harmless in-bounds data that lands in unread D columns, so no
        // EXEC predication is needed. Min/max go through an LDS walk
        // (v_max3/min3-fused by the compiler).
        float nsv, sum1, sum2;
#if __has_builtin(__builtin_amdgcn_wmma_f32_16x16x4_f32)
        {
            int cidx = lane & 15;
            int krow = (lane < 16) ? 0 : 2;
            int wsel = (cidx >= 3) ? 4 : 0;
            int st   = cidx % 3;
            float b0 = stats[krow + wsel][st];
            float b1 = stats[krow + 1 + wsel][st];
            v2f A;  A[0] = 1.0f; A[1] = 1.0f;
            v2f Bm; Bm[0] = b0;  Bm[1] = b1;
            v8f Cz = {};
            v8f D = __builtin_amdgcn_wmma_f32_16x16x4_f32(
                        false, A, false, Bm, (short)0, Cz, false, false);
            float d0 = D[0];  // row M=0: lanes 0..15 hold columns 0..15
            nsv  = readlane_f(d0, 0) + readlane_f(d0, 3);
            sum1 = readlane_f(d0, 1) + readlane_f(d0, 4);
            sum2 = readlane_f(d0, 2) + readlane_f(d0, 5);
        }
#else
        {
            nsv = 0.f; sum1 = 0.f; sum2 = 0.f;
            for (int wi = 0; wi < NW; ++wi) {
                nsv += stats[wi][0]; sum1 += stats[wi][1]; sum2 += stats[wi][2];
            }
        }
#endif
        float Bmx = -INFINITY, Bmn = INFINITY, Cmx = -INFINITY, Cmn = INFINITY;
        for (int wi = 0; wi < NW; ++wi) {
            Bmx = fmaxf(Bmx, stats[wi][3]); Bmn = fminf(Bmn, stats[wi][4]);
            Cmx = fmaxf(Cmx, stats[wi][5]); Cmn = fminf(Cmn, stats[wi][6]);
        }

        // scalar statistics with fast rcp/sqrt (feed only bucket thresholds)
        float n    = nsv * 4.0f;
        float mean = sum1 * FAST_RCP(fmaxf(n, 1.0f));
        float M2   = sum2 - 2.0f * mean * sum1 + mean * mean * n;
        float var  = fmaxf(M2, 0.0f) * FAST_RCP(fmaxf(n - 1.0f, 1.0f));
        float stdv = FAST_SQRT(var);
        // branchless 3x3 weight table
        float twc0 = (mean <= 0.2f) ? 0.9f : ((mean <= 0.5f) ? 0.6f : 0.4f);
        float twc1 = (mean <= 0.2f) ? 0.8f : ((mean <= 0.5f) ? 0.5f : 0.2f);
        float twc2 = (mean <= 0.2f) ? 0.6f : ((mean <= 0.5f) ? 0.4f : 0.1f);
        float tw   = (stdv <= 0.1f) ? twc0 : ((stdv <= 0.3f) ? twc1 : twc2);
        float Bd = Bmx - Bmn, Cd = Cmx - Cmn;
        float rBd = (Bd != 0.0f) ? FAST_RCP(Bd) : 0.0f;
        float rCd = (Cd != 0.0f) ? FAST_RCP(Cd) : 0.0f;
        float wB = tw + 0.3f, wC = 1.0f - tw + 0.55f;

        // per-lane best (strict < keeps lowest server index on ties)
        float bsc  = INFINITY;
        int   bidx = tid * SPT;
#pragma unroll
        for (int j = 0; j < SPT; ++j) {
            if (vld[j]) {
                float bi = (Bv[j] - Bmn) * rBd;
                float ci = (Cv[j] - Cmn) * rCd;
                float sc = wB * bi + wC * ci;     // >= 0 for valid servers
                if (sc < bsc) { bsc = sc; bidx = tid * SPT + j; }
            }
        }
        // wave argmin: f32 min tree + ballot of achievers + readlane of index
        float wmin = bsc;
        for (int m = 16; m > 0; m >>= 1) wmin = fminf(wmin, __shfl_xor(wmin, m));
        unsigned em = (unsigned)__ballot(bsc == wmin);
        int srclane = __builtin_ctz(em);          // lowest achieving lane
        unsigned widx = (unsigned)readlane_i(bidx, srclane);
        if (lane == 0)
            minbuf[wv] = (((unsigned long long)__float_as_uint(wmin)) << 32) | widx;

        // drain async copy before publishing the next mask buffer
        asm volatile("s_wait_asynccnt 0" ::: "memory");
        __syncthreads();  // barrier 2

        unsigned long long gbest = minbuf[0];
        for (int wi = 1; wi < NW; ++wi) {
            unsigned long long o = minbuf[wi];
            gbest = (o < gbest) ? o : gbest;
        }
        unsigned chosen = (unsigned)(gbest & 0xffffffffu);
        bool any = ((unsigned)(gbest >> 32)) != 0x7f800000u;

        // owner thread applies the allocation and refreshes cached stats
        if (any && (chosen >> 2) == (unsigned)tid) {
            int j = chosen & 3;
            tmp[j][0] -= w0; tmp[j][1] -= w1; tmp[j][2] -= w2; tmp[j][3] -= w3;
            usage[j] += 1.0f;
            float fs = 0.f, fq = 0.f, bs = 0.f;
#pragma unroll
            for (int c = 0; c < 4; ++c) {
                float x = (cap[j][c] - tmp[j][c]) * rcap[j][c];
                float f;
                if (x <= 0.0f)            f = 0.0f;
                else if (x <= mids[c])    f = x * rmid[c];
                else if (x <= highs[c])   f = (highs[c] - x) * rspan[c];
                else                      f = 0.0f;
                fs += f; fq += f * f; bs += tmp[j][c] * rcap[j][c];
            }
            fsum[j] = fs; fsq[j] = fq; Bv[j] = 1.0f - 0.25f * bs; Cv[j] = 0.0f;
        }
        if (tid == 0) {
            out[ouid] = any ? (float)chosen : -1.0f;
            allocCount += any ? 1 : 0;
        }
    }

    // ---------- phase 4: outputs ----------
    __syncthreads();
#pragma unroll
    for (int j = 0; j < SPT; ++j) out[U + tid * SPT + j] = usage[j];
    float su = 0.f;
#pragma unroll
    for (int j = 0; j < SPT; ++j) su += (usage[j] != 0.0f) ? 1.0f : 0.0f;
    for (int m = 16; m > 0; m >>= 1) su += __shfl_xor(su, m);
    if (lane == 0) stats[wv][0] = su;
    __syncthreads();
    if (tid == 0) {
        float stotal = 0.f;
        for (int wi = 0; wi < NW; ++wi) stotal += stats[wi][0];
        out[U + S]     = (float)allocCount / (float)U;
        out[U + S + 1] = stotal / (float)S;
    }
}

extern "C" void kernel_launch(void* const* d_in, const int* in_sizes, int n_in,
                              void* d_out, int out_size, void* d_ws, size_t ws_size,
                              hipStream_t stream) {
    (void)n_in; (void)out_size; (void)ws_size;
    const float* servers        = (const float*)d_in[0];
    const float* users          = (const float*)d_in[1];
    const unsigned char* masks  = (const unsigned char*)d_in[2];  // jax bool = 1 byte
    int S = in_sizes[0] / 7;   // 1024
    int U = in_sizes[1] / 6;   // 8192
    fuzzy_alloc_kernel<<<1, NTH, 0, stream>>>(
        servers, users, masks, (float*)d_out,
        (unsigned long long*)d_ws, U, S);
}